// PersonalizedAttentionModel_22213570854958
// MI455X (gfx1250) — compile-verified
//
#include <hip/hip_runtime.h>

typedef __attribute__((ext_vector_type(2))) float v2f;
typedef __attribute__((ext_vector_type(8))) float v8f;
typedef __attribute__((ext_vector_type(4))) float f4;

#define NB   1024   // batch
#define DIN  128
#define HID  256
#define DOUT 64
#define NSUB 64
#define NT   256    // threads per block = 8 wave32s

#define PR     8                 // k-rows per weight slice
#define SSW    48                // slice row stride (floats): all B reads bank-conflict-free
#define SLICEF (PR * SSW)        // floats per slice buffer (384)
#define NRING  4                 // slice ring buffers per wave (prefetch depth 3)

// ---- CDNA5 async copy global->LDS (ASYNCcnt path) --------------------------
__device__ __forceinline__ unsigned lds_addr(const void* p) {
    // generic pointer to __shared__ = {SHARED_BASE, offset}; LDS aperture truncates
    // to addr[31:0], so the low 32 bits are the DS byte address.
    return (unsigned)(uintptr_t)p;
}
// GVS form: 64-bit uniform base in SGPR pair + 32-bit per-lane byte offset VGPR.
__device__ __forceinline__ void async_copy_b128(unsigned lds_byte, unsigned goff,
                                                const void* base) {
    asm volatile("global_load_async_to_lds_b128 %0, %1, %2"
                 :: "v"(lds_byte), "v"(goff), "s"(base) : "memory");
}
template <int CNT>
__device__ __forceinline__ void wait_async() {
    // async loads complete in order: waiting for <= CNT outstanding guarantees all
    // but the CNT most recently issued copies have landed in LDS.
    asm volatile("s_wait_asynccnt %0" :: "i"(CNT) : "memory");
}

__device__ __forceinline__ v8f wmma4(v2f a, v2f b, v8f c) {
    return __builtin_amdgcn_wmma_f32_16x16x4_f32(false, a, false, b, (short)0, c,
                                                 false, false);
}

// One 16-row M-tile through one dense layer: dst[16xN] = src[16xK] @ W[KxN] + bias.
// Each wave owns up to two 16x16 N-tiles; its B columns are private, so it streams
// them through a private 4-buffer LDS ring (3 panels prefetched) with per-wave
// ASYNCcnt sync only (no block barriers inside the layer). All guards are scalar
// (readfirstlane) so EXEC is all-ones at every WMMA.
template <int K, int N, int SSTR, int DSTR>
__device__ __forceinline__ void layer_tc(
    const float* __restrict__ src, float* __restrict__ dst,
    float (* __restrict__ sl)[NRING][SLICEF],
    const float* __restrict__ W, const float* __restrict__ bias, int tid)
{
    const int lane  = tid & 31;
    const int swave = __builtin_amdgcn_readfirstlane(tid >> 5);   // uniform
    const int half  = lane >> 4;
    const int l16   = lane & 15;
    const int kofs  = half << 1;
    constexpr int NTILES = N >> 4;
    constexpr int P      = K / PR;             // 16 or 32 (multiple of 4)
    constexpr int IFL    = (N == HID) ? 2 : 1; // async ops per slice

    if (swave >= NTILES) return;               // scalar branch; layer has no barriers
    const int nt0 = swave;                     // second tile (swave+8) exists iff N==HID
    float* s0 = sl[swave][0];
    float* s1 = sl[swave][1];
    float* s2 = sl[swave][2];
    float* s3 = sl[swave][3];

    // issue async copies of this wave's B slice for panel p into 'dslice'
    auto issue = [&](int p, float* dslice) {
        if constexpr (N == HID) {
            #pragma unroll
            for (int j = 0; j < 2; ++j) {      // 64 f4: 8 rows x 2 tiles x 4 f4
                const int e    = j * 32 + lane;
                const int r    = e >> 3;
                const int cc   = e & 7;
                const int tile = cc >> 2;
                const int c4   = (cc & 3) << 2;
                const int nt   = tile ? (swave + 8) : swave;
                const unsigned goff =
                    (unsigned)(((p * PR + r) * N + (nt << 4) + c4) * 4);
                async_copy_b128(lds_addr(dslice + r * SSW + (tile << 4) + c4), goff, W);
            }
        } else {                               // N==DOUT: 32 f4: 8 rows x 1 tile x 4 f4
            const int r  = lane >> 2;
            const int c4 = (lane & 3) << 2;
            const unsigned goff =
                (unsigned)(((p * PR + r) * N + (nt0 << 4) + c4) * 4);
            async_copy_b128(lds_addr(dslice + r * SSW + c4), goff, W);
        }
    };

    v8f acc0 = {}, acc1 = {};
    auto step = [&](const float* cs, int p) {  // consume slice for panel p
        const float* Ac = src + l16 * SSTR + p * PR + kofs;
        #pragma unroll
        for (int kb = 0; kb < PR / 4; ++kb) {
            v2f a;
            a.x = Ac[(kb << 2)];
            a.y = Ac[(kb << 2) + 1];
            const int kr = (kb << 2) + kofs;   // slice-local K row
            v2f b0;
            b0.x = cs[kr * SSW + l16];
            b0.y = cs[(kr + 1) * SSW + l16];
            acc0 = wmma4(a, b0, acc0);
            if constexpr (N == HID) {
                v2f b1;
                b1.x = cs[kr * SSW + 16 + l16];
                b1.y = cs[(kr + 1) * SSW + 16 + l16];
                acc1 = wmma4(a, b1, acc1);
            }
        }
    };

    // 4-deep ring, 3 panels prefetched; async completes in order so waiting for
    // <= n*IFL outstanding proves all but the newest n slices have landed.
    issue(0, s0); issue(1, s1); issue(2, s2);
    int p = 0;
    for (; p + 4 < P; p += 4) {
        issue(p + 3, s3); wait_async<3 * IFL>(); step(s0, p);
        issue(p + 4, s0); wait_async<3 * IFL>(); step(s1, p + 1);
        issue(p + 5, s1); wait_async<3 * IFL>(); step(s2, p + 2);
        issue(p + 6, s2); wait_async<3 * IFL>(); step(s3, p + 3);
    }
    // tail: p == P - 4
    issue(P - 1, s3);
    wait_async<3 * IFL>(); step(s0, P - 4);
    wait_async<2 * IFL>(); step(s1, P - 3);
    wait_async<1 * IFL>(); step(s2, P - 2);
    wait_async<0>();       step(s3, P - 1);

    // D layout: VGPR j, lane half h -> row M = j + 8h, col = l16
    {
        const float bn = bias[(nt0 << 4) + l16];
        float* Dc = dst + (size_t)(half << 3) * DSTR + (nt0 << 4) + l16;
        #pragma unroll
        for (int j = 0; j < 8; ++j) Dc[(size_t)j * DSTR] = acc0[j] + bn;
    }
    if constexpr (N == HID) {
        const float bn = bias[((swave + 8) << 4) + l16];
        float* Dc = dst + (size_t)(half << 3) * DSTR + ((swave + 8) << 4) + l16;
        #pragma unroll
        for (int j = 0; j < 8; ++j) Dc[(size_t)j * DSTR] = acc1[j] + bn;
    }
}

// ---------------- Backend: 4 shared dense layers, chained in LDS ----------------
__global__ __launch_bounds__(NT)
void backend_kernel(const float* __restrict__ x,
                    const float* __restrict__ Wb1, const float* __restrict__ bb1,
                    const float* __restrict__ Wbv, const float* __restrict__ bbv,
                    const float* __restrict__ Wbo, const float* __restrict__ bbo,
                    const float* __restrict__ Wb2, const float* __restrict__ bb2,
                    float* __restrict__ emb)
{
    __shared__ float bufA[16 * (HID + 4)];
    __shared__ float bufB[16 * (HID + 4)];
    __shared__ float sl[8][NRING][SLICEF];
    const int tid = threadIdx.x;
    const int m0  = blockIdx.x * 16;

    for (int i = tid; i < 16 * (DIN / 4); i += NT) {
        const int r = i / (DIN / 4), c = i % (DIN / 4);
        ((f4*)(bufA + r * (DIN + 4)))[c] = ((const f4*)(x + (size_t)(m0 + r) * DIN))[c];
    }
    __syncthreads();
    layer_tc<DIN, HID, DIN + 4, HID + 4>(bufA, bufB, sl, Wb1, bb1, tid);
    __syncthreads();
    layer_tc<HID, HID, HID + 4, HID + 4>(bufB, bufA, sl, Wbv, bbv, tid);
    __syncthreads();
    layer_tc<HID, HID, HID + 4, HID + 4>(bufA, bufB, sl, Wbo, bbo, tid);
    __syncthreads();
    layer_tc<HID, HID, HID + 4, HID + 4>(bufB, bufA, sl, Wb2, bb2, tid);
    __syncthreads();
    for (int i = tid; i < 16 * (HID / 4); i += NT) {
        const int r = i / (HID / 4), c = i % (HID / 4);
        ((f4*)(emb + (size_t)(m0 + r) * HID))[c] = ((const f4*)(bufA + r * (HID + 4)))[c];
    }
}

// ------------- Group samples by subject (offsets + gathered index list) -------------
__global__ __launch_bounds__(NB)
void group_kernel(const int* __restrict__ subject,
                  int* __restrict__ offsets,   // [NSUB+1]
                  int* __restrict__ idx)       // [NB]
{
    __shared__ int cnt[NSUB];
    __shared__ int base[NSUB];
    const int tid = threadIdx.x;
    if (tid < NSUB) cnt[tid] = 0;
    __syncthreads();
    const int s = subject[tid];
    atomicAdd(&cnt[s], 1);
    __syncthreads();
    if (tid == 0) {
        int run = 0;
        for (int i = 0; i < NSUB; ++i) { base[i] = run; offsets[i] = run; run += cnt[i]; }
        offsets[NSUB] = run;
    }
    __syncthreads();
    const int p = atomicAdd(&base[s], 1);
    idx[p] = tid;
}

// ---------------- Frontend: per-subject 4-layer expert chain ----------------
__global__ __launch_bounds__(NT)
void frontend_kernel(const float* __restrict__ emb,
                     const int* __restrict__ offsets, const int* __restrict__ idx,
                     const float* __restrict__ Wf1, const float* __restrict__ bf1,
                     const float* __restrict__ Wfv, const float* __restrict__ bfv,
                     const float* __restrict__ Wfo, const float* __restrict__ bfo,
                     const float* __restrict__ Wf2, const float* __restrict__ bf2,
                     float* __restrict__ out)
{
    const int s   = blockIdx.y;
    const int t   = blockIdx.x;
    const int beg = offsets[s];
    const int cnt = offsets[s + 1] - beg;
    if (t * 16 >= cnt) return;          // uniform early exit (before any barrier)

    __shared__ float bufA[16 * (HID + 4)];
    __shared__ float bufB[16 * (HID + 4)];
    __shared__ float sl[8][NRING][SLICEF];
    __shared__ int   rows[16];

    const int tid = threadIdx.x;
    if (tid < 16) {
        const int j = t * 16 + tid;
        rows[tid] = (j < cnt) ? idx[beg + j] : -1;   // padded rows computed, never stored
    }
    __syncthreads();
    for (int i = tid; i < 16 * (HID / 4); i += NT) {
        const int r = i / (HID / 4), c = i % (HID / 4);
        const int g = (rows[r] >= 0) ? rows[r] : 0;
        ((f4*)(bufA + r * (HID + 4)))[c] = ((const f4*)(emb + (size_t)g * HID))[c];
    }
    __syncthreads();

    const size_t wofs = (size_t)s * HID * HID;
    layer_tc<HID, HID, HID + 4, HID + 4>(bufA, bufB, sl, Wf1 + wofs, bf1 + s * HID, tid);
    __syncthreads();
    layer_tc<HID, HID, HID + 4, HID + 4>(bufB, bufA, sl, Wfv + wofs, bfv + s * HID, tid);
    __syncthreads();
    layer_tc<HID, HID, HID + 4, HID + 4>(bufA, bufB, sl, Wfo + wofs, bfo + s * HID, tid);
    __syncthreads();
    layer_tc<HID, DOUT, HID + 4, DOUT + 4>(bufB, bufA, sl,
                                           Wf2 + (size_t)s * HID * DOUT, bf2 + s * DOUT, tid);
    __syncthreads();

    // masked scatter: 16 rows x 64 floats (16 f4) -> out[rows[r]]
    const int r = tid >> 4, c = tid & 15;
    if (rows[r] >= 0)
        ((f4*)(out + (size_t)rows[r] * DOUT))[c] = ((const f4*)(bufA + r * (DOUT + 4)))[c];
}

extern "C" void kernel_launch(void* const* d_in, const int* in_sizes, int n_in,
                              void* d_out, int out_size, void* d_ws, size_t ws_size,
                              hipStream_t stream) {
    const float* features = (const float*)d_in[0];
    const int*   subject  = (const int*)  d_in[1];
    const float* Wb1 = (const float*)d_in[2];  const float* bb1 = (const float*)d_in[3];
    const float* Wbv = (const float*)d_in[4];  const float* bbv = (const float*)d_in[5];
    const float* Wbo = (const float*)d_in[6];  const float* bbo = (const float*)d_in[7];
    const float* Wb2 = (const float*)d_in[8];  const float* bb2 = (const float*)d_in[9];
    const float* Wf1 = (const float*)d_in[10]; const float* bf1 = (const float*)d_in[11];
    const float* Wfv = (const float*)d_in[12]; const float* bfv = (const float*)d_in[13];
    const float* Wfo = (const float*)d_in[14]; const float* bfo = (const float*)d_in[15];
    const float* Wf2 = (const float*)d_in[16]; const float* bf2 = (const float*)d_in[17];
    float* out = (float*)d_out;

    // workspace: emb[1024*256] f32, offsets[65] i32, idx[1024] i32
    char* ws = (char*)d_ws;
    float* emb     = (float*)ws;
    int*   offsets = (int*)(ws + (size_t)NB * HID * sizeof(float));
    int*   idx     = offsets + (NSUB + 1);

    backend_kernel<<<dim3(NB / 16), NT, 0, stream>>>(
        features, Wb1, bb1, Wbv, bbv, Wbo, bbo, Wb2, bb2, emb);
    group_kernel<<<1, NB, 0, stream>>>(subject, offsets, idx);
    frontend_kernel<<<dim3(NB / 16, NSUB), NT, 0, stream>>>(
        emb, offsets, idx, Wf1, bf1, Wfv, bfv, Wfo, bfo, Wf2, bf2, out);
}